// VQEmbedding_24721831756116
// MI455X (gfx1250) — compile-verified
//
#include <hip/hip_runtime.h>

typedef __attribute__((ext_vector_type(2))) float v2f;
typedef __attribute__((ext_vector_type(8))) float v8f;

#define K_CODES 8192
#define DIM 512
#define N_ROWS 16384
#define LDS_STRIDE 516   // 512 + 4 pad floats: kills LDS bank conflicts on strided b64 reads
#define ROWS_PER_BLOCK 64

// ---------------------------------------------------------------------------
// Kernel A: enorm[k] = ||codebook[k]||^2   (one wave per codebook row)
// ---------------------------------------------------------------------------
__global__ void __launch_bounds__(256) cb_norms(const float* __restrict__ cb,
                                                float* __restrict__ enorm) {
    const int wave = threadIdx.x >> 5;
    const int lane = threadIdx.x & 31;
    const int row  = blockIdx.x * 8 + wave;
    const float4* p = (const float4*)(cb + (size_t)row * DIM + lane * 16);
    float s = 0.f;
#pragma unroll
    for (int i = 0; i < 4; ++i) {
        float4 v = p[i];
        s += v.x * v.x + v.y * v.y + v.z * v.z + v.w * v.w;
    }
#pragma unroll
    for (int m = 16; m >= 1; m >>= 1) s += __shfl_xor(s, m, 32);
    if (lane == 0) enorm[row] = s;
}

// ---------------------------------------------------------------------------
// Kernel B: fused GEMM (-2 * z @ cb^T via V_WMMA_F32_16X16X4_F32) + argmin.
// Block = 256 threads = 8 waves, owns 64 z-rows (4 row-tiles per wave).
// Wave w scans codebook columns [w*1024, (w+1)*1024), 4 col-tiles at a time:
// per k-step, 4 global b64 B-frags + 4 LDS b64 A-frags feed 16 WMMAs.
// ---------------------------------------------------------------------------
__global__ void __launch_bounds__(256) vq_argmin(const float* __restrict__ z,
                                                 const float* __restrict__ cb,
                                                 const float* __restrict__ enorm,
                                                 int* __restrict__ idx_out) {
    __shared__ float As[ROWS_PER_BLOCK * LDS_STRIDE];   // -2 * z tile, padded (~129 KB)
    __shared__ float red_val[8 * ROWS_PER_BLOCK];
    __shared__ int   red_idx[8 * ROWS_PER_BLOCK];

    const int tid     = threadIdx.x;
    const int wave    = tid >> 5;
    const int lane    = tid & 31;
    const int rowbase = blockIdx.x * ROWS_PER_BLOCK;

    // Stage A = -2*z for 64 rows into LDS (coalesced float4 reads).
    {
        const float4* zsrc = (const float4*)(z + (size_t)rowbase * DIM);
        for (int c = tid; c < ROWS_PER_BLOCK * DIM / 4; c += 256) {
            float4 v = zsrc[c];
            int r = c >> 7;                 // / (512/4)
            int col = (c & 127) << 2;
            float* d = &As[r * LDS_STRIDE + col];
            d[0] = -2.f * v.x; d[1] = -2.f * v.y;
            d[2] = -2.f * v.z; d[3] = -2.f * v.w;
        }
    }
    __syncthreads();

    const int lclass = lane & 15;           // col-within-tile / row M class
    const int koff   = (lane >> 4) << 1;    // lanes 0-15: K{0,1}; 16-31: K{2,3}
    const v2f* aptr0 = (const v2f*)(&As[(lclass +  0) * LDS_STRIDE + koff]);
    const v2f* aptr1 = (const v2f*)(&As[(lclass + 16) * LDS_STRIDE + koff]);
    const v2f* aptr2 = (const v2f*)(&As[(lclass + 32) * LDS_STRIDE + koff]);
    const v2f* aptr3 = (const v2f*)(&As[(lclass + 48) * LDS_STRIDE + koff]);

    float rmin[4][8];
    int   ridx[4][8];
#pragma unroll
    for (int rt = 0; rt < 4; ++rt)
#pragma unroll
        for (int v = 0; v < 8; ++v) { rmin[rt][v] = 3.4e38f; ridx[rt][v] = 0; }

    const int colw = wave * 1024;           // this wave's column range
    for (int q = 0; q < 16; ++q) {
        const int base = colw + q * 64;
        const v2f* b0 = (const v2f*)(cb + (size_t)(base +  0 + lclass) * DIM + koff);
        const v2f* b1 = (const v2f*)(cb + (size_t)(base + 16 + lclass) * DIM + koff);
        const v2f* b2 = (const v2f*)(cb + (size_t)(base + 32 + lclass) * DIM + koff);
        const v2f* b3 = (const v2f*)(cb + (size_t)(base + 48 + lclass) * DIM + koff);

        v8f acc[4][4];
#pragma unroll
        for (int rt = 0; rt < 4; ++rt)
#pragma unroll
            for (int t = 0; t < 4; ++t) acc[rt][t] = (v8f){};

#pragma unroll 1
        for (int k = 0; k < DIM / 4; ++k) {  // 128 k-steps of 4
            v2f a0  = aptr0[k * 2];          // ds_load_b64 x4 (row-tiles)
            v2f a1  = aptr1[k * 2];
            v2f a2  = aptr2[k * 2];
            v2f a3  = aptr3[k * 2];
            v2f vb0 = b0[k * 2];             // global_load_b64 x4 (L2-resident)
            v2f vb1 = b1[k * 2];
            v2f vb2 = b2[k * 2];
            v2f vb3 = b3[k * 2];
            acc[0][0] = __builtin_amdgcn_wmma_f32_16x16x4_f32(false, a0, false, vb0,
                                                              (short)0, acc[0][0], false, false);
            acc[0][1] = __builtin_amdgcn_wmma_f32_16x16x4_f32(false, a0, false, vb1,
                                                              (short)0, acc[0][1], false, false);
            acc[0][2] = __builtin_amdgcn_wmma_f32_16x16x4_f32(false, a0, false, vb2,
                                                              (short)0, acc[0][2], false, false);
            acc[0][3] = __builtin_amdgcn_wmma_f32_16x16x4_f32(false, a0, false, vb3,
                                                              (short)0, acc[0][3], false, false);
            acc[1][0] = __builtin_amdgcn_wmma_f32_16x16x4_f32(false, a1, false, vb0,
                                                              (short)0, acc[1][0], false, false);
            acc[1][1] = __builtin_amdgcn_wmma_f32_16x16x4_f32(false, a1, false, vb1,
                                                              (short)0, acc[1][1], false, false);
            acc[1][2] = __builtin_amdgcn_wmma_f32_16x16x4_f32(false, a1, false, vb2,
                                                              (short)0, acc[1][2], false, false);
            acc[1][3] = __builtin_amdgcn_wmma_f32_16x16x4_f32(false, a1, false, vb3,
                                                              (short)0, acc[1][3], false, false);
            acc[2][0] = __builtin_amdgcn_wmma_f32_16x16x4_f32(false, a2, false, vb0,
                                                              (short)0, acc[2][0], false, false);
            acc[2][1] = __builtin_amdgcn_wmma_f32_16x16x4_f32(false, a2, false, vb1,
                                                              (short)0, acc[2][1], false, false);
            acc[2][2] = __builtin_amdgcn_wmma_f32_16x16x4_f32(false, a2, false, vb2,
                                                              (short)0, acc[2][2], false, false);
            acc[2][3] = __builtin_amdgcn_wmma_f32_16x16x4_f32(false, a2, false, vb3,
                                                              (short)0, acc[2][3], false, false);
            acc[3][0] = __builtin_amdgcn_wmma_f32_16x16x4_f32(false, a3, false, vb0,
                                                              (short)0, acc[3][0], false, false);
            acc[3][1] = __builtin_amdgcn_wmma_f32_16x16x4_f32(false, a3, false, vb1,
                                                              (short)0, acc[3][1], false, false);
            acc[3][2] = __builtin_amdgcn_wmma_f32_16x16x4_f32(false, a3, false, vb2,
                                                              (short)0, acc[3][2], false, false);
            acc[3][3] = __builtin_amdgcn_wmma_f32_16x16x4_f32(false, a3, false, vb3,
                                                              (short)0, acc[3][3], false, false);
        }

        // dist = ||e||^2 - 2 z.e  (the -2 z.e already sits in acc)
#pragma unroll
        for (int t = 0; t < 4; ++t) {
            const int col = base + t * 16 + lclass;
            const float en = enorm[col];
#pragma unroll
            for (int rt = 0; rt < 4; ++rt) {
#pragma unroll
                for (int v = 0; v < 8; ++v) {
                    float d = en + acc[rt][t][v];
                    if (d < rmin[rt][v]) { rmin[rt][v] = d; ridx[rt][v] = col; }  // strict <: keeps first
                }
            }
        }
    }

    // Cross-lane (16 column classes) min+argmin, half-wave groups stay separate.
#pragma unroll
    for (int rt = 0; rt < 4; ++rt) {
#pragma unroll
        for (int v = 0; v < 8; ++v) {
            float mv = rmin[rt][v];
            int   mi = ridx[rt][v];
#pragma unroll
            for (int m = 8; m >= 1; m >>= 1) {
                float ov = __shfl_xor(mv, m, 16);
                int   oi = __shfl_xor(mi, m, 16);
                if (ov < mv || (ov == mv && oi < mi)) { mv = ov; mi = oi; }
            }
            if (lclass == 0) {
                int r = rt * 16 + ((lane >> 4) << 3) + v;  // 16 rows per row-tile
                red_val[wave * ROWS_PER_BLOCK + r] = mv;
                red_idx[wave * ROWS_PER_BLOCK + r] = mi;
            }
        }
    }
    __syncthreads();

    // Cross-wave reduce (waves scan ascending column ranges -> strict < keeps first).
    if (tid < ROWS_PER_BLOCK) {
        float best = red_val[tid];
        int   bi   = red_idx[tid];
        for (int w = 1; w < 8; ++w) {
            float v = red_val[w * ROWS_PER_BLOCK + tid];
            int   i = red_idx[w * ROWS_PER_BLOCK + tid];
            if (v < best || (v == best && i < bi)) { best = v; bi = i; }
        }
        idx_out[rowbase + tid] = bi;
    }
}

// ---------------------------------------------------------------------------
// Kernel C: z_st = z + (codebook[idx] - z); deterministic per-block partial
// sums of (zq - z)^2.
// ---------------------------------------------------------------------------
__global__ void __launch_bounds__(256) vq_gather(const float* __restrict__ z,
                                                 const float* __restrict__ cb,
                                                 const int* __restrict__ idx,
                                                 float* __restrict__ out,
                                                 float* __restrict__ partials) {
    __shared__ float sdata[256];
    const int tid = threadIdx.x;
    const size_t base = (size_t)blockIdx.x * 8192;
    float acc = 0.f;
#pragma unroll 4
    for (int i = 0; i < 32; ++i) {
        size_t e = base + (size_t)i * 256 + tid;
        int row = (int)(e >> 9);
        int col = (int)(e & 511);
        float zq = cb[(size_t)idx[row] * DIM + col];
        float zv = z[e];
        float d  = zq - zv;
        out[e] = zv + d;                // matches z + stop_grad(zq - z) bit-for-bit
        acc += d * d;
    }
    sdata[tid] = acc;
    __syncthreads();
    for (int s = 128; s > 0; s >>= 1) {
        if (tid < s) sdata[tid] += sdata[tid + s];
        __syncthreads();
    }
    if (tid == 0) partials[blockIdx.x] = sdata[0];
}

// ---------------------------------------------------------------------------
// Kernel D: final fixed-order reduction -> vq_loss = 1.1 * mean
// ---------------------------------------------------------------------------
__global__ void __launch_bounds__(256) vq_loss(const float* __restrict__ partials,
                                               float* __restrict__ out_loss) {
    __shared__ float sdata[256];
    const int tid = threadIdx.x;
    float a = partials[tid] + partials[tid + 256] +
              partials[tid + 512] + partials[tid + 768];
    sdata[tid] = a;
    __syncthreads();
    for (int s = 128; s > 0; s >>= 1) {
        if (tid < s) sdata[tid] += sdata[tid + s];
        __syncthreads();
    }
    if (tid == 0)
        out_loss[0] = sdata[0] * (1.1f / ((float)N_ROWS * (float)DIM));
}

// ---------------------------------------------------------------------------
extern "C" void kernel_launch(void* const* d_in, const int* in_sizes, int n_in,
                              void* d_out, int out_size, void* d_ws, size_t ws_size,
                              hipStream_t stream) {
    const float* z  = (const float*)d_in[0];   // [16,1024,512]
    const float* cb = (const float*)d_in[1];   // [8192,512]
    float* out = (float*)d_out;                // [16*1024*512] z_st + [1] loss

    float* enorm    = (float*)d_ws;            // 8192 f32
    int*   idx      = (int*)(enorm + K_CODES); // 16384 i32
    float* partials = (float*)(idx + N_ROWS);  // 1024 f32

    cb_norms <<<K_CODES / 8, 256, 0, stream>>>(cb, enorm);
    vq_argmin<<<N_ROWS / ROWS_PER_BLOCK, 256, 0, stream>>>(z, cb, enorm, idx);
    vq_gather<<<1024, 256, 0, stream>>>(z, cb, idx, out, partials);
    vq_loss  <<<1, 256, 0, stream>>>(partials, out + (size_t)N_ROWS * DIM);
}